// MultiHeadAttention_61890478735373
// MI455X (gfx1250) — compile-verified
//
#include <hip/hip_runtime.h>
#include <math.h>

typedef __attribute__((ext_vector_type(2))) float v2f;
typedef __attribute__((ext_vector_type(8))) float v8f;

#define WMMA_F32X4(a, b, c) \
  __builtin_amdgcn_wmma_f32_16x16x4_f32(false, (a), false, (b), (short)0, (c), false, false)

// A/B fragment loader for NT GEMM with row-major source (ld = K).
// f32 16x4 A-layout: lanes 0-15 -> row (row0+lane), K = k..k+1 ;
//                    lanes 16-31 -> row (row0+lane-16), K = k+2..k+3.
static __device__ __forceinline__ v2f frag_nt(const float* base, int ld,
                                              int row0, int k, int lane) {
  const float* p = base + (size_t)(row0 + (lane & 15)) * ld + k + ((lane >> 4) << 1);
  return *(const v2f*)p;  // 8B aligned: ld even, k multiple of 4
}

// Generic pointer to a __shared__ object: low 32 bits are the LDS byte offset.
static __device__ __forceinline__ uint32_t lds_off(const void* p) {
  return (uint32_t)(uintptr_t)p;
}

// CDNA5 async DMA: 16 bytes global -> LDS per lane, tracked by ASYNCcnt.
static __device__ __forceinline__ void async_b128(const float* g, uint32_t lds_byte) {
  asm volatile("global_load_async_to_lds_b128 %0, %1, off"
               :: "v"(lds_byte), "v"(g) : "memory");
}

// ---------------------------------------------------------------------------
// GEMM C[M,N] = A[M,K] * W[N,K]^T + bias[N]
// 256 threads (8 waves); block tile 128 rows x 128 cols; wave tile 16x128.
// W tile (128x32) double-buffered in LDS via async copies.
// mode 0: scatter into Q (B,H,T,Dh), K (B,H,T,Dh), V^T (B,H,Dh,T)
// mode 1: plain row-major output
// ---------------------------------------------------------------------------
#define GBK  32
#define GLDS 44   // padded stride (floats): 176B rows (16B aligned), conflict-free

__global__ __launch_bounds__(256) void gemm_nt_kernel(
    const float* __restrict__ A, const float* __restrict__ W,
    const float* __restrict__ bias,
    float* __restrict__ q_out, float* __restrict__ k_out,
    float* __restrict__ vt_out, float* __restrict__ plain_out,
    int M, int N, int K, int mode) {
  __shared__ float Bs[2][128 * GLDS];
  const int tid  = threadIdx.x;
  const int lane = tid & 31;
  const int wave = tid >> 5;
  const int col0 = blockIdx.x * 128;
  const int row0 = blockIdx.y * 128 + wave * 16;

  v8f acc[8];
  v8f zero = {};
#pragma unroll
  for (int i = 0; i < 8; ++i) acc[i] = zero;

  // per-thread staging slice: 16 consecutive floats of one W row
  const int scol = tid >> 1;           // 0..127
  const int skb  = (tid & 1) << 4;     // 0 or 16
  const float* sbase = W + (size_t)(col0 + scol) * K + skb;
  const uint32_t sdst[2] = { lds_off(&Bs[0][scol * GLDS + skb]),
                             lds_off(&Bs[1][scol * GLDS + skb]) };

  auto stage = [&](int kc, int buf) {
    const float* s = sbase + kc;
#pragma unroll
    for (int j = 0; j < 4; ++j) async_b128(s + 4 * j, sdst[buf] + 16 * j);
  };

  stage(0, 0);
  const int nch = K / GBK;
  for (int ch = 0; ch < nch; ++ch) {
    const int kc = ch * GBK;
    if (ch + 1 < nch) {
      stage(kc + GBK, (ch + 1) & 1);
      asm volatile("s_wait_asynccnt 0x4" ::: "memory");  // prev chunk's 4 done
    } else {
      asm volatile("s_wait_asynccnt 0x0" ::: "memory");
    }
    __syncthreads();
    const float* bb = Bs[ch & 1];
#pragma unroll
    for (int kk = 0; kk < GBK; kk += 4) {
      v2f a = frag_nt(A, K, row0, kc + kk, lane);
      // batch all 8 B-fragment loads first: distinct dest regs -> the ds
      // loads issue as a clause and overlap the WMMA stream.
      v2f bfr[8];
#pragma unroll
      for (int nt = 0; nt < 8; ++nt)
        bfr[nt] = *(const v2f*)&bb[(nt * 16 + (lane & 15)) * GLDS + kk + ((lane >> 4) << 1)];
#pragma unroll
      for (int nt = 0; nt < 8; ++nt)
        acc[nt] = WMMA_F32X4(a, bfr[nt], acc[nt]);
    }
    __syncthreads();
  }

  // epilogue: VGPR r holds row (row0 + r + 8*(lane>=16)), col = lane&15
#pragma unroll
  for (int nt = 0; nt < 8; ++nt) {
    const int col = col0 + nt * 16 + (lane & 15);
    const float bv = bias[col];
#pragma unroll
    for (int r = 0; r < 8; ++r) {
      const int row = row0 + r + ((lane >> 4) << 3);
      const float v = acc[nt][r] + bv;
      if (mode == 0) {
        const int which = col >> 11;        // 0=q 1=k 2=v
        const int rem = col & 2047;
        const int h = rem >> 7, d = rem & 127;
        const int b = row >> 11, t = row & 2047;
        const size_t bh = (size_t)(b * 16 + h);
        if (which == 0)      q_out[(bh * 2048 + t) * 128 + d] = v;
        else if (which == 1) k_out[(bh * 2048 + t) * 128 + d] = v;
        else                 vt_out[(bh * 128 + d) * 2048 + t] = v;  // transposed V
      } else {
        plain_out[(size_t)row * N + col] = v;
      }
    }
  }
}

// ---------------------------------------------------------------------------
// Flash attention: 256 threads = 8 waves; 8 | 128 so all 8 waves handle
// consecutive 16-row q tiles of the SAME (b,h). K/V k-tiles staged in LDS
// once per block (async, double-buffered), shared by 8 waves -> 8x less L2
// traffic. V is stored transposed (Dh,T) so PV is an NT GEMM too.
// ---------------------------------------------------------------------------
#define KLDS 132  // 16 x 128 K tile, padded row stride (528B, 16B aligned)
#define VLDS 20   // 128 x 16 V^T tile, padded row stride (80B, 16B aligned)

__global__ __launch_bounds__(256) void flash_attn_kernel(
    const float* __restrict__ Q, const float* __restrict__ Km,
    const float* __restrict__ Vt, float* __restrict__ O) {
  __shared__ float Ks[2][16 * KLDS];
  __shared__ float Vs[2][128 * VLDS];
  __shared__ float Ps[8][16 * 18];
  const int tid  = threadIdx.x;
  const int lane = tid & 31;
  const int wv   = tid >> 5;
  const int gw   = blockIdx.x * 8 + wv;   // 8 | 128 => all 8 waves share bh
  const int qt   = gw & 127;
  const int bh   = gw >> 7;               // 0..63
  const float* q  = Q  + (size_t)bh * 2048 * 128;
  const float* k  = Km + (size_t)bh * 2048 * 128;
  const float* vt = Vt + (size_t)bh * 128 * 2048;
  float* pls = Ps[wv];

  // staging slices (per thread, 8 floats for K tile + 8 floats for V tile)
  const int krow = tid >> 4, kcol = (tid & 15) << 3;   // K: 16 x 128
  const int vrow = tid >> 1, vcb  = (tid & 1) << 3;    // V^T: 128 x 16
  const float* ksrc = k + krow * 128 + kcol;
  const float* vsrc = vt + (size_t)vrow * 2048 + vcb;
  const uint32_t kdst[2] = { lds_off(&Ks[0][krow * KLDS + kcol]),
                             lds_off(&Ks[1][krow * KLDS + kcol]) };
  const uint32_t vdst[2] = { lds_off(&Vs[0][vrow * VLDS + vcb]),
                             lds_off(&Vs[1][vrow * VLDS + vcb]) };

  auto stage = [&](int kt, int buf) {
    const float* ks = ksrc + kt * 2048;   // rows kt*16 .. +15 of K
    const float* vs = vsrc + kt * 16;     // cols kt*16 .. +15 of V^T
    async_b128(ks,     kdst[buf]);
    async_b128(ks + 4, kdst[buf] + 16);
    async_b128(vs,     vdst[buf]);
    async_b128(vs + 4, vdst[buf] + 16);
  };

  // cache all Q fragments for this 16-row tile (K-depth 128 -> 32 frags)
  v2f qa[32];
#pragma unroll
  for (int i = 0; i < 32; ++i) qa[i] = frag_nt(q, 128, qt * 16, i * 4, lane);

  v8f o[8];
  v8f zero = {};
#pragma unroll
  for (int i = 0; i < 8; ++i) o[i] = zero;
  float m[8], l[8];
#pragma unroll
  for (int r = 0; r < 8; ++r) { m[r] = -1e30f; l[r] = 0.0f; }

  const float scale = 0.08838834764831845f;  // 1/sqrt(128)

  stage(0, 0);
  for (int kt = 0; kt < 128; ++kt) {
    const int buf = kt & 1;
    if (kt + 1 < 128) {
      stage(kt + 1, buf ^ 1);
      asm volatile("s_wait_asynccnt 0x4" ::: "memory");  // prev tile's 4 done
    } else {
      asm volatile("s_wait_asynccnt 0x0" ::: "memory");
    }
    __syncthreads();

    // S = Q * K^T  (16x16 tile, 32 wmma), K fragments from LDS in batches of 4
    v8f s = zero;
#pragma unroll
    for (int i4 = 0; i4 < 32; i4 += 4) {
      v2f bfr[4];
#pragma unroll
      for (int j = 0; j < 4; ++j)
        bfr[j] = *(const v2f*)&Ks[buf][(lane & 15) * KLDS + (i4 + j) * 4 + ((lane >> 4) << 1)];
#pragma unroll
      for (int j = 0; j < 4; ++j)
        s = WMMA_F32X4(qa[i4 + j], bfr[j], s);
    }

    // online softmax: row r lives across lanes 0-15 (rows 0-7) / 16-31 (8-15)
    float pr[8];
#pragma unroll
    for (int r = 0; r < 8; ++r) {
      float sv = s[r] * scale;
      float mx = sv;
      mx = fmaxf(mx, __shfl_xor(mx, 1, 32));
      mx = fmaxf(mx, __shfl_xor(mx, 2, 32));
      mx = fmaxf(mx, __shfl_xor(mx, 4, 32));
      mx = fmaxf(mx, __shfl_xor(mx, 8, 32));
      const float mn = fmaxf(m[r], mx);
      const float alpha = __expf(m[r] - mn);
      m[r] = mn;
      const float p = __expf(sv - mn);
      float rs = p;
      rs += __shfl_xor(rs, 1, 32);
      rs += __shfl_xor(rs, 2, 32);
      rs += __shfl_xor(rs, 4, 32);
      rs += __shfl_xor(rs, 8, 32);
      l[r] = l[r] * alpha + rs;
#pragma unroll
      for (int nt = 0; nt < 8; ++nt) o[nt][r] *= alpha;
      pr[r] = p;
    }

    // C-layout -> A-layout for P via padded LDS (cross-lane: explicit wait)
#pragma unroll
    for (int r = 0; r < 8; ++r)
      pls[(r + ((lane >> 4) << 3)) * 18 + (lane & 15)] = pr[r];
    asm volatile("s_wait_dscnt 0x0" ::: "memory");
    v2f pa[4];
#pragma unroll
    for (int i = 0; i < 4; ++i)
      pa[i] = *(const v2f*)&pls[(lane & 15) * 18 + i * 4 + ((lane >> 4) << 1)];

    // O += P * V : V^T fragments from LDS (rows = head dims), batched loads
#pragma unroll
    for (int nt = 0; nt < 8; ++nt) {
      v2f bfr[4];
#pragma unroll
      for (int i = 0; i < 4; ++i)
        bfr[i] = *(const v2f*)&Vs[buf][(nt * 16 + (lane & 15)) * VLDS + i * 4 + ((lane >> 4) << 1)];
#pragma unroll
      for (int i = 0; i < 4; ++i)
        o[nt] = WMMA_F32X4(pa[i], bfr[i], o[nt]);
    }
    __syncthreads();
  }

  // normalize and write to (B, T, D) so out-proj reads it row-major
  float inv[8];
#pragma unroll
  for (int r = 0; r < 8; ++r) inv[r] = 1.0f / l[r];
  const int b = bh >> 4, h = bh & 15;
#pragma unroll
  for (int nt = 0; nt < 8; ++nt) {
#pragma unroll
    for (int r = 0; r < 8; ++r) {
      const int row = qt * 16 + r + ((lane >> 4) << 3);
      const int col = h * 128 + nt * 16 + (lane & 15);
      O[((size_t)b * 2048 + row) * 2048 + col] = o[nt][r] * inv[r];
    }
  }
}

// ---------------------------------------------------------------------------
extern "C" void kernel_launch(void* const* d_in, const int* in_sizes, int n_in,
                              void* d_out, int out_size, void* d_ws, size_t ws_size,
                              hipStream_t stream) {
  const float* x     = (const float*)d_in[0];  // (4,2048,2048)
  const float* qkv_w = (const float*)d_in[1];  // (6144,2048)
  const float* qkv_b = (const float*)d_in[2];  // (6144)
  const float* out_w = (const float*)d_in[3];  // (2048,2048)
  const float* out_b = (const float*)d_in[4];  // (2048)
  float* out = (float*)d_out;                  // (4,2048,2048)

  const size_t SEG = (size_t)4 * 16 * 2048 * 128;  // 16.7M floats = 64 MB
  float* qbuf  = (float*)d_ws;       // (B,H,T,Dh)
  float* kbuf  = qbuf  + SEG;        // (B,H,T,Dh)
  float* vtbuf = kbuf  + SEG;        // (B,H,Dh,T)
  float* abuf  = vtbuf + SEG;        // (B,T,D) attention output

  // QKV projection: (8192,2048) x (6144,2048)^T, scatter into q/k/vt
  gemm_nt_kernel<<<dim3(48, 64), 256, 0, stream>>>(
      x, qkv_w, qkv_b, qbuf, kbuf, vtbuf, nullptr, 8192, 6144, 2048, 0);

  // attention: 8192 waves, 8 per block (same (b,h) per block)
  flash_attn_kernel<<<1024, 256, 0, stream>>>(qbuf, kbuf, vtbuf, abuf);

  // output projection: (8192,2048) x (2048,2048)^T + bias
  gemm_nt_kernel<<<dim3(16, 64), 256, 0, stream>>>(
      abuf, out_w, out_b, nullptr, nullptr, nullptr, out, 8192, 2048, 2048, 1);
}